// Detector_20650202759314
// MI455X (gfx1250) — compile-verified
//
#include <hip/hip_runtime.h>
#include <stdint.h>

#define IOU_T 0.5f
#define TOPK 1024

// candidate counts: B=32, 3 anchors per cell
#define N13 16224    // 32*13*13*3
#define N26 64896    // 32*26*26*3
#define NTOT 340704  // + 32*52*52*3

typedef __attribute__((ext_vector_type(2))) float v2f;
typedef __attribute__((ext_vector_type(8))) float v8f;
typedef unsigned int u32;
typedef unsigned long long u64;
typedef unsigned short u16;

struct SelState {
  u64 prefix;
  u32 krem;
  u32 pad0;
  u64 cutoff;
  u32 counter;
  u32 pad1;
};

// ---------------- workspace layout ----------------
constexpr size_t WS_KEYS  = 0;                          // NTOT u64
constexpr size_t WS_HIST  = (size_t)NTOT * 8;           // 8*256 u32
constexpr size_t WS_STATE = WS_HIST + 2048 * 4;
constexpr size_t WS_CAND  = WS_STATE + 256;             // TOPK u64
constexpr size_t WS_BOXF  = WS_CAND + (size_t)TOPK * 8; // TOPK*7 f32
constexpr size_t WS_X1    = WS_BOXF + (size_t)TOPK * 7 * 4;
constexpr size_t WS_Y1    = WS_X1 + (size_t)TOPK * 4;
constexpr size_t WS_X2    = WS_Y1 + (size_t)TOPK * 4;
constexpr size_t WS_Y2    = WS_X2 + (size_t)TOPK * 4;
constexpr size_t WS_AREA  = WS_Y2 + (size_t)TOPK * 4;
constexpr size_t WS_SCORE = WS_AREA + (size_t)TOPK * 4;
constexpr size_t WS_M     = WS_SCORE + (size_t)TOPK * 4; // 1024*64 u16 = 128 KB

// ---------------- helpers ----------------
struct Loc { const float* base; int GG; int G; int gx; int gy; int b; int a; };

__device__ __forceinline__ Loc locate(int i, const float* in13, const float* in26,
                                      const float* in52) {
  Loc L;
  const float* in;
  int G;
  if (i < N13)            { in = in13; G = 13; }
  else if (i < N13 + N26) { in = in26; G = 26; i -= N13; }
  else                    { in = in52; G = 52; i -= (N13 + N26); }
  int a = i % 3; int cell = i / 3;
  int gx = cell % G; int cy = cell / G;
  int gy = cy % G;   int b  = cy / G;
  L.G = G; L.GG = G * G; L.gx = gx; L.gy = gy; L.b = b; L.a = a;
  // layout (B,255,G,G); channel for anchor a, field c is a*85+c
  L.base = in + (((size_t)(b * 255 + a * 85) * G + gy) * G + gx);
  return L;
}

__device__ __forceinline__ float sigm(float x) { return 1.0f / (1.0f + expf(-x)); }

// ---------------- kernels ----------------
__global__ void k_init(u32* hist, SelState* st) {
  int tid = threadIdx.x;
  for (int i = tid; i < 2048; i += 256) hist[i] = 0;
  if (tid == 0) { st->prefix = 0ull; st->krem = TOPK; st->cutoff = 0ull; st->counter = 0u; }
}

// score pass: reads only the conf-logit channel (1.4 MB), builds order keys
__global__ void k_score(const float* in13, const float* in26, const float* in52,
                        const float* threshP, u64* keys) {
  int i = blockIdx.x * blockDim.x + threadIdx.x;
  if (i >= NTOT) return;
  Loc L = locate(i, in13, in26, in52);
  float conf = sigm(L.base[0]);
  float thresh = threshP[0];
  float score = (conf > thresh) ? conf : -1.0f;
  u32 s = __float_as_uint(score);
  u32 ord = (s & 0x80000000u) ? ~s : (s | 0x80000000u);  // ascending-float order
  u32 d = ~ord;                                          // descending-float order
  keys[i] = ((u64)d << 32) | (u32)i;                     // unique total order
}

// radix-select: histogram of digit p among keys matching the decided prefix
__global__ void k_hist(const u64* __restrict__ keys, const SelState* __restrict__ st,
                       u32* __restrict__ hist, int p) {
  int i = blockIdx.x * blockDim.x + threadIdx.x;
  if (i >= NTOT) return;
  if (i + 8192 < NTOT) __builtin_prefetch(&keys[i + 8192], 0, 0);  // global_prefetch_b8
  u64 key = keys[i];
  int shift = 56 - 8 * p;
  bool match = true;
  if (p > 0) match = (key >> (shift + 8)) == (st->prefix >> (shift + 8));
  if (match) atomicAdd(&hist[(p << 8) + (int)((key >> shift) & 0xFF)], 1u);
}

__global__ void k_pick(SelState* st, const u32* hist, int p) {
  int shift = 56 - 8 * p;
  u32 k = st->krem;
  u32 c = 0;
  const u32* h = hist + (p << 8);
  for (int d = 0; d < 256; ++d) {
    u32 hd = h[d];
    if (c + hd >= k) { st->prefix |= ((u64)d) << shift; st->krem = k - c; break; }
    c += hd;
  }
  if (p == 7) st->cutoff = st->prefix;  // exact 1024th-smallest key (keys unique)
}

__global__ void k_compact(const u64* keys, SelState* st, u64* cand) {
  int i = blockIdx.x * blockDim.x + threadIdx.x;
  if (i >= NTOT) return;
  u64 key = keys[i];
  if (key <= st->cutoff) {
    u32 pos = atomicAdd(&st->counter, 1u);
    if (pos < TOPK) cand[pos] = key;
  }
}

// one block: bitonic-sort 1024 keys in LDS, decode full boxes for winners only
__global__ void __launch_bounds__(1024)
k_sort_decode(const u64* cand, const float* in13, const float* in26, const float* in52,
              const float* caseP, const float* a13, const float* a26, const float* a52,
              float* boxf, float* X1, float* Y1, float* X2, float* Y2,
              float* AREA, float* SCORE) {
  __shared__ u64 skey[TOPK];
  int tid = threadIdx.x;
  skey[tid] = cand[tid];
  __syncthreads();
  for (int k = 2; k <= TOPK; k <<= 1)
    for (int j = k >> 1; j > 0; j >>= 1) {
      int ix = tid ^ j;
      if (ix > tid) {
        u64 va = skey[tid], vb = skey[ix];
        bool up = ((tid & k) == 0);
        if ((va > vb) == up) { skey[tid] = vb; skey[ix] = va; }
      }
      __syncthreads();
    }
  u64 key = skey[tid];
  u32 idx = (u32)key;
  // recover the exact score float from the key (bijective transform)
  u32 dd = (u32)(key >> 32);
  u32 u = ~dd;
  u32 fb = (u & 0x80000000u) ? (u ^ 0x80000000u) : ~u;
  float score = __uint_as_float(fb);

  Loc L = locate((int)idx, in13, in26, in52);
  const float* anch = (idx < N13) ? a13 : ((idx < N13 + N26) ? a26 : a52);
  float cse = caseP[0];
  float t = 416.0f / (float)L.G;
  int GG = L.GG;
  float conf = sigm(L.base[0]);
  float o2 = L.base[2 * GG];
  float o3 = L.base[3 * GG];
  float o4 = L.base[4 * GG];
  float cy = ((float)L.gy + o2) * t / cse;   // reference uses out[...,2] for BOTH
  float cx = ((float)L.gx + o2) * t / cse;
  float w = anch[L.a * 2 + 0] * expf(o3) / cse;
  float h = anch[L.a * 2 + 1] * expf(o4) / cse;
  float best = L.base[5 * GG];
  int bc = 0;
  for (int c = 1; c < 80; ++c) {               // argmax(softmax) == argmax(logits)
    float v = L.base[(5 + c) * GG];
    if (v > best) { best = v; bc = c; }
  }
  float* bp = boxf + tid * 7;
  bp[0] = (float)L.b; bp[1] = conf; bp[2] = cx; bp[3] = cy;
  bp[4] = w; bp[5] = h; bp[6] = (float)bc;
  X1[tid] = cx - w * 0.5f; Y1[tid] = cy - h * 0.5f;
  X2[tid] = cx + w * 0.5f; Y2[tid] = cy + h * 0.5f;
  AREA[tid] = w * h; SCORE[tid] = score;
}

// pairwise IoU suppression matrix: one 16x16 tile per wave.
// area[i]+area[j] computed on the matrix pipe with v_wmma_f32_16x16x4_f32:
// A = [area_i, 1, 0, 0] (16x4), B = [1, area_j, 0, 0]^T (4x16) -> C = outer sum.
__global__ void k_nms_matrix(const float* __restrict__ X1, const float* __restrict__ Y1,
                             const float* __restrict__ X2, const float* __restrict__ Y2,
                             const float* __restrict__ AREA, u16* __restrict__ M) {
  __shared__ u32 rowbits[8][16];
  int wave = threadIdx.x >> 5;
  int lane = threadIdx.x & 31;
  int tile = blockIdx.x * 8 + wave;  // 512 blocks * 8 waves = 4096 tiles (64x64)
  int ti = tile >> 6, tj = tile & 63;

  if (lane < 16) rowbits[wave][lane] = 0;

  v2f A, B;
  if (lane < 16) {  // A lanes 0-15: K=0 (area_i), K=1 (1.0); lanes 16-31: K=2,3 = 0
    A.x = AREA[ti * 16 + lane]; A.y = 1.0f;
    B.x = 1.0f;                 B.y = AREA[tj * 16 + lane];
  } else {
    A.x = 0.0f; A.y = 0.0f; B.x = 0.0f; B.y = 0.0f;
  }
  v8f C = {};
  v8f D = __builtin_amdgcn_wmma_f32_16x16x4_f32(false, A, false, B, (short)0, C,
                                                false, false);
  // C/D layout: lane l, vgpr v -> row M = v + 8*(l>=16), col N = l%16
  int j = tj * 16 + (lane & 15);
  float x1j = X1[j], y1j = Y1[j], x2j = X2[j], y2j = Y2[j];
  int mhi = (lane >> 4) * 8;
#pragma unroll
  for (int v = 0; v < 8; ++v) {
    int m = v + mhi;
    int i = ti * 16 + m;
    float x1i = X1[i], y1i = Y1[i], x2i = X2[i], y2i = Y2[i];
    float iw = fmaxf(fminf(x2i, x2j) - fmaxf(x1i, x1j), 0.0f);
    float ih = fmaxf(fminf(y2i, y2j) - fmaxf(y1i, y1j), 0.0f);
    float inter = iw * ih;
    float iou = inter / (D[v] - inter + 1e-9f);  // D[v] = area_i + area_j (WMMA)
    if ((iou > IOU_T) && (j > i)) atomicOr(&rowbits[wave][m], 1u << (lane & 15));
  }
  if (lane < 16) M[(size_t)(ti * 16 + lane) * 64 + tj] = (u16)rowbits[wave][lane];
}

// sequential suppression over the LDS-resident 128KB bit-matrix (fits in 320KB WGP LDS)
__global__ void __launch_bounds__(1024)
k_final(const u16* Mg, const float* SCORE, const float* boxf, float* out) {
  extern __shared__ u32 sm[];
  u32* ldsM = sm;          // 32768 words = 128 KB
  u32* supp = sm + 32768;  // 32 words = 1024 bits
  const u32* Mg32 = (const u32*)Mg;
  int tid = threadIdx.x;
  for (int w = tid; w < 32768; w += 1024) ldsM[w] = Mg32[w];
  if (tid < 32) supp[tid] = 0;
  __syncthreads();
  if (SCORE[tid] <= 0.0f) atomicOr(&supp[tid >> 5], 1u << (tid & 31));
  __syncthreads();
  if (tid < 32) {  // wave 0 only: dependent loop, wave-synchronous, LDS in-order
    volatile u32* vs = supp;
    for (int i = 0; i < TOPK; ++i) {
      u32 sw = vs[i >> 5];
      if (!((sw >> (i & 31)) & 1u)) vs[tid] |= ldsM[i * 32 + tid];
    }
  }
  __syncthreads();
  bool keep = !((supp[tid >> 5] >> (tid & 31)) & 1u);
  float kf = keep ? 1.0f : 0.0f;
  for (int c = 0; c < 7; ++c) out[tid * 7 + c] = boxf[tid * 7 + c] * kf;
}

// ---------------- host ----------------
extern "C" void kernel_launch(void* const* d_in, const int* in_sizes, int n_in,
                              void* d_out, int out_size, void* d_ws, size_t ws_size,
                              hipStream_t stream) {
  const float* in13 = (const float*)d_in[0];
  const float* in26 = (const float*)d_in[1];
  const float* in52 = (const float*)d_in[2];
  const float* threshP = (const float*)d_in[3];
  const float* caseP = (const float*)d_in[4];
  const float* a13 = (const float*)d_in[5];
  const float* a26 = (const float*)d_in[6];
  const float* a52 = (const float*)d_in[7];

  char* ws = (char*)d_ws;
  u64* keys = (u64*)(ws + WS_KEYS);
  u32* hist = (u32*)(ws + WS_HIST);
  SelState* st = (SelState*)(ws + WS_STATE);
  u64* cand = (u64*)(ws + WS_CAND);
  float* boxf = (float*)(ws + WS_BOXF);
  float* X1 = (float*)(ws + WS_X1);
  float* Y1 = (float*)(ws + WS_Y1);
  float* X2 = (float*)(ws + WS_X2);
  float* Y2 = (float*)(ws + WS_Y2);
  float* AREA = (float*)(ws + WS_AREA);
  float* SCORE = (float*)(ws + WS_SCORE);
  u16* M = (u16*)(ws + WS_M);
  float* out = (float*)d_out;

  int gN = (NTOT + 255) / 256;
  k_init<<<1, 256, 0, stream>>>(hist, st);
  k_score<<<gN, 256, 0, stream>>>(in13, in26, in52, threshP, keys);
  for (int p = 0; p < 8; ++p) {
    k_hist<<<gN, 256, 0, stream>>>(keys, st, hist, p);
    k_pick<<<1, 1, 0, stream>>>(st, hist, p);
  }
  k_compact<<<gN, 256, 0, stream>>>(keys, st, cand);
  k_sort_decode<<<1, 1024, 0, stream>>>(cand, in13, in26, in52, caseP, a13, a26, a52,
                                        boxf, X1, Y1, X2, Y2, AREA, SCORE);
  k_nms_matrix<<<512, 256, 0, stream>>>(X1, Y1, X2, Y2, AREA, M);
  k_final<<<1, 1024, (32768 + 32) * 4, stream>>>(M, SCORE, boxf, out);
}